// GNNModel_7550552507207
// MI455X (gfx1250) — compile-verified
//
#include <hip/hip_runtime.h>
#include <stdint.h>

typedef __attribute__((ext_vector_type(2))) float v2f;
typedef __attribute__((ext_vector_type(8))) float v8f;

#define FDIM 16

// ---- Pass 1: degree (self-loop baked in as init=1.0) ----
__global__ void gcn_deg_init(float* __restrict__ deg, int N) {
    int i = blockIdx.x * blockDim.x + threadIdx.x;
    if (i < N) deg[i] = 1.0f;
}

__global__ void gcn_deg_accum(const long long* __restrict__ ei,
                              float* __restrict__ deg, int E) {
    int e = blockIdx.x * blockDim.x + threadIdx.x;
    if (e < E) {
        int dst = (int)ei[(long long)E + e];
        atomicAdd(&deg[dst], 1.0f);
    }
}

__global__ void gcn_dinv(float* __restrict__ deg, int N) {
    int i = blockIdx.x * blockDim.x + threadIdx.x;
    if (i < N) {
        float d = deg[i];
        deg[i] = (d > 0.0f) ? rsqrtf(d) : 0.0f;   // overwrite in place
    }
}

// ---- Pass 2: xw = x @ W1 via V_WMMA_F32_16X16X4_F32 (K=2 zero-padded to 4).
// Pre-scale by dinv[src]; init h1 with the self-loop term + bias.
// One wave32 handles a 16-node x 16-feature tile. N is a multiple of 16, so
// no per-lane tail guards: all selects are register cndmasks, no exec churn.
__global__ void gcn_l1_wmma(const float* __restrict__ x,
                            const float* __restrict__ W1,
                            const float* __restrict__ b1,
                            const float* __restrict__ dinv,
                            float* __restrict__ s1,
                            float* __restrict__ h1, int N) {
    int lane = threadIdx.x & 31;
    int wave = threadIdx.x >> 5;
    int tile = blockIdx.x * (blockDim.x >> 5) + wave;
    int base = tile * 16;
    if (base >= N) return;                 // wave-uniform: EXEC all-1s for WMMA

    int col = lane & 15;
    bool lo = lane < 16;                   // lanes 16..31 carry K=2,3 (zero padding)

    // Unconditional loads; addresses valid for every lane (col-based).
    int mrd = base + col;
    float ax = x[mrd * 2 + 0];
    float ay = x[mrd * 2 + 1];
    float bx = W1[0 * 16 + col];
    float by = W1[1 * 16 + col];

    // A: 16x4 tile of x. Lanes 0-15: VGPR0=K0, VGPR1=K1; lanes 16-31: K2,K3 = 0.
    v2f a; a.x = lo ? ax : 0.0f; a.y = lo ? ay : 0.0f;
    // B: W1 padded 2x16 -> 4x16. Lanes 0-15: rows K0,K1; lanes 16-31: K2,K3 = 0.
    v2f b; b.x = lo ? bx : 0.0f; b.y = lo ? by : 0.0f;

    v8f c = {};
    // 8 args: (neg_a, A, neg_b, B, c_mod, C, reuse_a, reuse_b)
    v8f d = __builtin_amdgcn_wmma_f32_16x16x4_f32(
        false, a, false, b, (short)0, c, false, false);

    float bias = b1[col];
    // D layout: VGPR r -> lanes 0-15: M=r, lanes 16-31: M=r+8; N=col.
    int mbase = base + (lo ? 0 : 8);

    float di[8];
    #pragma unroll
    for (int r = 0; r < 8; ++r) di[r] = dinv[mbase + r];   // batched loads

    #pragma unroll
    for (int r = 0; r < 8; ++r) {
        int m = mbase + r;
        float sv = d[r] * di[r];                 // pre-scaled by dinv[src]
        s1[m * FDIM + col] = sv;
        h1[m * FDIM + col] = __builtin_fmaf(sv, di[r], bias);  // self-loop + bias
    }
}

// ---- Pass 3: layer-1 edge scatter. 16 lanes per edge, one feature per lane:
// coalesced 64B gather of s1[src], coalesced 16x global_atomic_add_f32 to h1[dst].
__global__ void gcn_l1_edges(const long long* __restrict__ ei,
                             const float* __restrict__ s1,
                             const float* __restrict__ dinv,
                             float* __restrict__ h1, int E) {
    int t = blockIdx.x * blockDim.x + threadIdx.x;
    int e = t >> 4;
    int f = t & 15;
    if (e >= E) return;
    if (e + 8192 < E) __builtin_prefetch(&ei[e + 8192], 0, 1);  // global_prefetch_b8
    int src = (int)ei[e];
    int dst = (int)ei[(long long)E + e];
    float v = s1[src * FDIM + f] * dinv[dst];
    atomicAdd(&h1[dst * FDIM + f], v);
}

// ---- Pass 4: layer-2 projection q = (h1 . W2) * dinv; init out with self term.
__global__ void gcn_l2_proj(const float* __restrict__ h1,
                            const float* __restrict__ W2,
                            const float* __restrict__ b2,
                            const float* __restrict__ dinv,
                            float* __restrict__ q,
                            float* __restrict__ out, int N) {
    int i = blockIdx.x * blockDim.x + threadIdx.x;
    if (i >= N) return;
    const float4* hp = (const float4*)(h1 + (size_t)i * FDIM);
    float p = 0.0f;
    #pragma unroll
    for (int k = 0; k < 4; ++k) {
        float4 hv = hp[k];
        p += hv.x * W2[4 * k + 0] + hv.y * W2[4 * k + 1] +
             hv.z * W2[4 * k + 2] + hv.w * W2[4 * k + 3];
    }
    float di = dinv[i];
    float qq = p * di;
    q[i] = qq;
    out[i] = qq * di + b2[0];              // self-loop + bias (also inits d_out)
}

// ---- Pass 5: layer-2 scalar edge scatter. ----
__global__ void gcn_l2_edges(const long long* __restrict__ ei,
                             const float* __restrict__ q,
                             const float* __restrict__ dinv,
                             float* __restrict__ out, int E) {
    int e = blockIdx.x * blockDim.x + threadIdx.x;
    if (e >= E) return;
    int src = (int)ei[e];
    int dst = (int)ei[(long long)E + e];
    atomicAdd(&out[dst], q[src] * dinv[dst]);
}

extern "C" void kernel_launch(void* const* d_in, const int* in_sizes, int n_in,
                              void* d_out, int out_size, void* d_ws, size_t ws_size,
                              hipStream_t stream) {
    const float*     x  = (const float*)d_in[0];
    const long long* ei = (const long long*)d_in[1];
    const float*     W1 = (const float*)d_in[2];
    const float*     b1 = (const float*)d_in[3];
    const float*     W2 = (const float*)d_in[4];
    const float*     b2 = (const float*)d_in[5];
    float* out = (float*)d_out;

    int N = in_sizes[0] / 2;   // x is [N,2]
    int E = in_sizes[1] / 2;   // edge_index is [2,E]

    // Workspace: dinv[N] | s1[N*16] | h1[N*16];  q reuses s1 after layer 1.
    float* dinv = (float*)d_ws;
    float* s1   = dinv + N;
    float* h1   = s1 + (size_t)N * FDIM;
    float* q    = s1;

    const int BLK = 256;
    int gN = (N + BLK - 1) / BLK;
    int gE = (E + BLK - 1) / BLK;
    int gT = (N + 127) / 128;                              // 8 waves/block * 16 nodes/wave
    long long tl1 = (long long)E * FDIM;
    int gL1 = (int)((tl1 + BLK - 1) / BLK);

    gcn_deg_init <<<gN,  BLK, 0, stream>>>(dinv, N);
    gcn_deg_accum<<<gE,  BLK, 0, stream>>>(ei, dinv, E);
    gcn_dinv     <<<gN,  BLK, 0, stream>>>(dinv, N);
    gcn_l1_wmma  <<<gT,  BLK, 0, stream>>>(x, W1, b1, dinv, s1, h1, N);
    gcn_l1_edges <<<gL1, BLK, 0, stream>>>(ei, s1, dinv, h1, E);
    gcn_l2_proj  <<<gN,  BLK, 0, stream>>>(h1, W2, b2, dinv, q, out, N);
    gcn_l2_edges <<<gE,  BLK, 0, stream>>>(ei, q, dinv, out, E);
}